// BlockAttentionResidual_1537598292480
// MI455X (gfx1250) — compile-verified
//
#include <hip/hip_runtime.h>
#include <math.h>

// MI455X (gfx1250) — bandwidth-bound routed-softmax kernel.
// 576 MB of traffic @ 23.3 TB/s => ~25us floor. Single-pass streaming:
// each source element is loaded exactly once (NT, 128-bit), kept in VGPRs,
// reused for both the score reduction and the weighted combine.
// Cross-wave reduction rides the WMMA pipe (v_wmma_f32_16x16x4_f32, A=ones).

typedef float v4f __attribute__((ext_vector_type(4)));
typedef float v2f __attribute__((ext_vector_type(2)));
typedef float v8f __attribute__((ext_vector_type(8)));

#define NSRC 8
#define DDIM 2048
#define TPB  256
#define EPT  (DDIM / TPB)   // 8 floats per thread per source row

__global__ __launch_bounds__(TPB)
void routed_softmax_kernel(const float* __restrict__ src,   // [BT, NSRC, DDIM]
                           const float* __restrict__ wq,    // [DDIM]
                           const float* __restrict__ nw,    // [DDIM]
                           float* __restrict__ out)         // [BT, DDIM]
{
    const int tok  = blockIdx.x;
    const int tid  = threadIdx.x;
    const int lane = tid & 31;
    const int wave = tid >> 5;

    const float* tbase = src + (size_t)tok * (NSRC * DDIM);
    const int dofs = tid * EPT;

    // Effective query = w_query * norm_weight (hot in L2: only 8 KB each).
    v4f q0 = *(const v4f*)(wq + dofs);
    v4f q1 = *(const v4f*)(wq + dofs + 4);
    q0 *= *(const v4f*)(nw + dofs);
    q1 *= *(const v4f*)(nw + dofs + 4);

    // Stream all 8 source rows into VGPRs, non-temporal (512MB > 192MB L2).
    v4f s[NSRC][2];
#pragma unroll
    for (int n = 0; n < NSRC; ++n) {
        const v4f* p = (const v4f*)(tbase + n * DDIM + dofs);
        s[n][0] = __builtin_nontemporal_load(p);
        s[n][1] = __builtin_nontemporal_load(p + 1);
    }

    // Per-thread partials: part[n] = sum s^2, part[8+n] = sum q*s.
    float part[16];
#pragma unroll
    for (int n = 0; n < NSRC; ++n) {
        v4f a = s[n][0], b = s[n][1];
        v4f sq = a * a + b * b;
        v4f dt = a * q0 + b * q1;
        part[n]     = (sq.x + sq.y) + (sq.z + sq.w);
        part[8 + n] = (dt.x + dt.y) + (dt.z + dt.w);
    }

    // wave32 butterfly reduction (all lanes end with the wave sum).
#pragma unroll
    for (int i = 0; i < 16; ++i) {
        float v = part[i];
        v += __shfl_xor(v, 16, 32);
        v += __shfl_xor(v,  8, 32);
        v += __shfl_xor(v,  4, 32);
        v += __shfl_xor(v,  2, 32);
        v += __shfl_xor(v,  1, 32);
        part[i] = v;
    }

    __shared__ float wred[8][16];   // [wave][quantity]
    if (lane == 0) {
#pragma unroll
        for (int i = 0; i < 16; ++i) wred[wave][i] = part[i];
    }
    __syncthreads();

    float tot[16];
#if __has_builtin(__builtin_amdgcn_wmma_f32_16x16x4_f32)
    {
        // Cross-wave reduction on the matrix pipe:
        //   D = ones(16x4) x B(4x16) + C  =>  column n of D = sum of B's 4 rows.
        // Two chained WMMAs fold all 8 wave partials for all 16 quantities.
        // A == all-ones makes the result invariant to the K-row lane mapping.
        const int qi = lane & 15;
        const int hi = lane >> 4;
        v2f a;  a.x = 1.0f; a.y = 1.0f;
        v2f b0, b1;
        b0.x = wred[2 * hi + 0][qi];
        b0.y = wred[2 * hi + 1][qi];
        b1.x = wred[2 * hi + 4][qi];
        b1.y = wred[2 * hi + 5][qi];
        v8f c = {};
        c = __builtin_amdgcn_wmma_f32_16x16x4_f32(false, a, false, b0, (short)0, c, false, false);
        c = __builtin_amdgcn_wmma_f32_16x16x4_f32(false, a, false, b1, (short)0, c, false, false);
        float t = c[0];                 // lane L holds total for quantity (L & 15)
#pragma unroll
        for (int i = 0; i < 16; ++i) tot[i] = __shfl(t, i, 32);
    }
#else
#pragma unroll
    for (int i = 0; i < 16; ++i) {
        float v = 0.0f;
#pragma unroll
        for (int w = 0; w < 8; ++w) v += wred[w][i];
        tot[i] = v;
    }
#endif

    // scores = (q . s) * rsqrt(mean(s^2)+eps) / sqrt(D); softmax over the 8 sources.
    const float inv_sqrt_d = 0.022097086912079612f;   // 1/sqrt(2048)
    float sc[NSRC];
    float m = -INFINITY;
#pragma unroll
    for (int n = 0; n < NSRC; ++n) {
        float inv_rms = rsqrtf(tot[n] * (1.0f / DDIM) + 1e-6f);
        sc[n] = tot[8 + n] * inv_rms * inv_sqrt_d;
        m = fmaxf(m, sc[n]);
    }
    float denom = 0.0f;
#pragma unroll
    for (int n = 0; n < NSRC; ++n) {
        sc[n] = __expf(sc[n] - m);
        denom += sc[n];
    }
    const float rdenom = 1.0f / denom;

    // routed[d] = sum_n w[n] * s[n][d]  (raw sources), NT 128-bit stores.
    v4f acc0 = {}, acc1 = {};
#pragma unroll
    for (int n = 0; n < NSRC; ++n) {
        float w = sc[n] * rdenom;
        acc0 += s[n][0] * w;
        acc1 += s[n][1] * w;
    }
    v4f* o = (v4f*)(out + (size_t)tok * DDIM + dofs);
    __builtin_nontemporal_store(acc0, o);
    __builtin_nontemporal_store(acc1, o + 1);
}

extern "C" void kernel_launch(void* const* d_in, const int* in_sizes, int n_in,
                              void* d_out, int out_size, void* d_ws, size_t ws_size,
                              hipStream_t stream)
{
    const float* src = (const float*)d_in[0];   // [B, T, N, D] fp32
    const float* wq  = (const float*)d_in[1];   // [D] fp32
    const float* nw  = (const float*)d_in[2];   // [D] fp32
    float* out = (float*)d_out;                 // [B, T, D] fp32

    const int BT = in_sizes[0] / (NSRC * DDIM); // 8192 tokens
    routed_softmax_kernel<<<BT, TPB, 0, stream>>>(src, wq, nw, out);
}